// RtReconstructionLayer_11897059410242
// MI455X (gfx1250) — compile-verified
//
#include <hip/hip_runtime.h>

typedef __attribute__((ext_vector_type(2))) float v2f;
typedef __attribute__((ext_vector_type(8))) float v8f;

#define WAVES_PER_BLOCK 8
#define BLOCK_THREADS (WAVES_PER_BLOCK * 32)
#define LDS_STRIDE 160  // floats per wave region (144 used, padded to 16B mult)

// F = K^-T * skew(t) * R(q) * K^-1 per row, via per-tile [16x12]x[12x16] fp32 WMMA:
//   F_flat(row) = G_flat(row) . W^T,  W[3i+m][3j+l] = K[3j+i]*K[3l+m],  G = skew(t)*R(q)
__global__ __launch_bounds__(BLOCK_THREADS)
void rt_reconstruction_wmma_kernel(const float* __restrict__ x,
                                   const float* __restrict__ Kinv,
                                   float* __restrict__ out,
                                   int N, int numTiles, int totalWaves) {
  __shared__ float lds[WAVES_PER_BLOCK * LDS_STRIDE];
  const int lane = threadIdx.x & 31;
  const int waveId = threadIdx.x >> 5;
  const int gwave = blockIdx.x * WAVES_PER_BLOCK + waveId;
  const int h = lane >> 4;   // lane half: selects K sub-pair per ISA A/B layout
  const int ln = lane & 15;  // row-in-tile (A) / output column n (B,D)
  float* wlds = lds + waveId * LDS_STRIDE;

  // ---- constant K^-1 (broadcast loads) ----
  float K[9];
#pragma unroll
  for (int i = 0; i < 9; ++i) K[i] = Kinv[i];

  // ---- B operand: WT[k][n] = K[3*(k/3)+n/3] * K[3*(k%3)+n%3]; zero-pad k>=9, n>=9
  // Layout: wmma i, vgpr v holds k = 4*i + 2*h + v at column n = ln.
  v2f b0, b1, b2;
  {
    const int n = ln;
    const int ni = n / 3, nm = n % 3;
    const int kb = 2 * h;
#define WT(k) (((n) < 9 && (k) < 9) ? K[3 * ((k) / 3) + ni] * K[3 * ((k) % 3) + nm] : 0.0f)
    b0[0] = WT(kb);     b0[1] = WT(kb + 1);
    b1[0] = WT(kb + 4); b1[1] = WT(kb + 5);
    b2[0] = WT(kb + 8); b2[1] = WT(kb + 9);
#undef WT
  }

  for (int t = gwave; t < numTiles; t += totalWaves) {  // wave-uniform loop
    const int base = t * 16;
    const int row = base + ln;
    const int rowc = row < N ? row : N - 1;  // clamp loads; stores are masked
    const float* p = x + (long long)rowc * 7;
    const float qw = p[0], qx = p[1], qy = p[2], qz = p[3];
    const float t0 = p[4], t1 = p[5], t2 = p[6];

    const float xx = qx * qx, yy = qy * qy, zz = qz * qz;
    const float xy = qx * qy, xz = qx * qz, yz = qy * qz;
    const float wx = qw * qx, wy = qw * qy, wz = qw * qz;

    const float r00 = 1.0f - 2.0f * (yy + zz);
    const float r01 = 2.0f * (xy - wz);
    const float r02 = 2.0f * (xz + wy);
    const float r10 = 2.0f * (xy + wz);
    const float r11 = 1.0f - 2.0f * (xx + zz);
    const float r12 = 2.0f * (yz - wx);
    const float r20 = 2.0f * (xz - wy);
    const float r21 = 2.0f * (yz + wx);
    const float r22 = 1.0f - 2.0f * (xx + yy);

    // G = skew(t) * R, row-major g0..g8
    const float g0 = -t2 * r10 + t1 * r20;
    const float g1 = -t2 * r11 + t1 * r21;
    const float g2 = -t2 * r12 + t1 * r22;
    const float g3 =  t2 * r00 - t0 * r20;
    const float g4 =  t2 * r01 - t0 * r21;
    const float g5 =  t2 * r02 - t0 * r22;
    const float g6 = -t1 * r00 + t0 * r10;
    const float g7 = -t1 * r01 + t0 * r11;
    const float g8 = -t1 * r02 + t0 * r12;

    // A operand (16x4 per wmma): lane half h holds k = 4*i + 2*h + {0,1}
    v2f a0, a1, a2;
    a0[0] = h ? g2 : g0;  a0[1] = h ? g3 : g1;
    a1[0] = h ? g6 : g4;  a1[1] = h ? g7 : g5;
    a2[0] = h ? 0.0f : g8; a2[1] = 0.0f;

    // ---- D[16x16] = A[16x12] * WT[12x16], fp32 WMMA chain (full EXEC here) ----
    v8f c = {};
    c = __builtin_amdgcn_wmma_f32_16x16x4_f32(false, a0, false, b0, (short)0, c,
                                              false, false);
    c = __builtin_amdgcn_wmma_f32_16x16x4_f32(false, a1, false, b1, (short)0, c,
                                              false, false);
    c = __builtin_amdgcn_wmma_f32_16x16x4_f32(false, a2, false, b2, (short)0, c,
                                              false, false);

    if (base + 16 <= N) {
      // D vgpr r -> (row M = r + 8*h, col n = ln); stage valid 16x9 to LDS,
      // then emit one coalesced b128 store + one b32 store (576 B contiguous).
      if (ln < 9) {
#pragma unroll
        for (int r = 0; r < 8; ++r) wlds[(r + 8 * h) * 9 + ln] = c[r];
      }
      asm volatile("s_wait_dscnt 0" ::: "memory");  // wave-local LDS RAW fence
      float* ob = out + (long long)base * 9;
      const float4 v4 = *(const float4*)(wlds + lane * 4);
      *(float4*)(ob + lane * 4) = v4;
      if (lane < 16) ob[128 + lane] = wlds[128 + lane];
    } else {
      // tail tile: masked scalar stores straight from D registers
#pragma unroll
      for (int r = 0; r < 8; ++r) {
        const int rw = base + r + 8 * h;
        if (rw < N && ln < 9) out[(long long)rw * 9 + ln] = c[r];
      }
    }
  }
}

extern "C" void kernel_launch(void* const* d_in, const int* in_sizes, int n_in,
                              void* d_out, int out_size, void* d_ws, size_t ws_size,
                              hipStream_t stream) {
  const float* x = (const float*)d_in[0];      // N x 7 f32
  const float* Kinv = (const float*)d_in[1];   // 3 x 3 f32
  float* out = (float*)d_out;                  // N x 9 f32

  const int N = in_sizes[0] / 7;
  if (N <= 0) return;
  const int numTiles = (N + 15) / 16;

  int blocks = (numTiles + WAVES_PER_BLOCK - 1) / WAVES_PER_BLOCK;
  if (blocks > 2048) blocks = 2048;  // grid-stride; ~16K waves saturates HBM
  const int totalWaves = blocks * WAVES_PER_BLOCK;

  rt_reconstruction_wmma_kernel<<<blocks, BLOCK_THREADS, 0, stream>>>(
      x, Kinv, out, N, numTiles, totalWaves);
}